// H2FDLayer_72748156059658
// MI455X (gfx1250) — compile-verified
//
#include <hip/hip_runtime.h>
#include <hip/hip_bf16.h>

// ---------------------------------------------------------------------------
// H2FD signed graph-attention layer, algebraically fused for MI455X (gfx1250).
//
// Node phase: single WMMA f32 GEMM [N x 64] @ [64 x 80] producing, per node:
//   cols 0..63 : hp = h @ W_w + b_w
//   col 64     : a_s = h . (W_d (Wf1+Wf3))
//   col 65     : a_d = h . (W_d (Wf2-Wf3))
//   cols 68..71: p[head] = hp_head . Wa[0:16]   (folded through W_w)
//   cols 72..75: q[head] = hp_head . Wa[16:32]
// Edge phase: sign = sign(a_s[src]+a_d[dst]+c); alpha = leaky_relu(sign*p+q+ba)
// then 3-pass segment softmax + weighted scatter (all L2-resident, 192MB L2).
// ---------------------------------------------------------------------------

typedef __attribute__((ext_vector_type(2))) float v2f;
typedef __attribute__((ext_vector_type(8))) float v8f;

#define NPACK_STRIDE 80

// ---------------- Precompute fused weights (tiny, 1 block) ------------------
__global__ __launch_bounds__(64) void precompute_kernel(
    const float* __restrict__ Ww, const float* __restrict__ bw,
    const float* __restrict__ Wa, const float* __restrict__ Wd,
    const float* __restrict__ bd, const float* __restrict__ Wf,
    const float* __restrict__ bf,
    float* __restrict__ wext, float* __restrict__ bext, float* __restrict__ cconst) {
  const int k = threadIdx.x;  // 0..63 : one row of W_ext
  float* row = wext + k * NPACK_STRIDE;
  for (int c = 0; c < 64; ++c) row[c] = Ww[k * 64 + c];
  float u = 0.f, v = 0.f;
  for (int j = 0; j < 64; ++j) {
    const float wd = Wd[k * 64 + j];
    u += wd * (Wf[j] + Wf[128 + j]);        // W_d (Wf1 + Wf3)
    v += wd * (Wf[64 + j] - Wf[128 + j]);   // W_d (Wf2 - Wf3)
  }
  row[64] = u; row[65] = v; row[66] = 0.f; row[67] = 0.f;
  for (int hd = 0; hd < 4; ++hd) {
    float p = 0.f, q = 0.f;
    for (int j = 0; j < 16; ++j) {
      const float ww = Ww[k * 64 + hd * 16 + j];
      p += ww * Wa[j];
      q += ww * Wa[16 + j];
    }
    row[68 + hd] = p;
    row[72 + hd] = q;
  }
  for (int c = 76; c < 80; ++c) row[c] = 0.f;

  if (k == 0) {
    for (int c = 0; c < 64; ++c) bext[c] = bw[c];
    bext[64] = 0.f; bext[65] = 0.f; bext[66] = 0.f; bext[67] = 0.f;
    for (int hd = 0; hd < 4; ++hd) {
      float p = 0.f, q = 0.f;
      for (int j = 0; j < 16; ++j) {
        p += bw[hd * 16 + j] * Wa[j];
        q += bw[hd * 16 + j] * Wa[16 + j];
      }
      bext[68 + hd] = p;
      bext[72 + hd] = q;
    }
    for (int c = 76; c < 80; ++c) bext[c] = 0.f;
    float cc = bf[0];
    for (int j = 0; j < 64; ++j) cc += bd[j] * (Wf[j] + Wf[64 + j]);
    cconst[0] = cc;
  }
}

// ---------------- Node GEMM via V_WMMA_F32_16X16X4_F32 ----------------------
// One wave per 16-node tile. 5 N-tiles of 16 cols, K=64 in 16 steps of 4.
// A (16x4 f32) layout: lanes 0-15 hold M=lane, K={0,1}; lanes 16-31 K={2,3}.
// B (4x16 f32)  layout: lanes 0-15 hold K={0,1}, N=lane; lanes 16-31 K={2,3}.
// C/D (16x16)   layout: VGPR r -> M=r (lanes 0-15) / M=r+8 (lanes 16-31).
__global__ __launch_bounds__(256) void node_gemm_kernel(
    const float* __restrict__ h, const float* __restrict__ wext,
    const float* __restrict__ bext, float* __restrict__ npack,
    int n_nodes, int n_tiles) {
  const int lane = threadIdx.x & 31;
  const int tile = blockIdx.x * 8 + (threadIdx.x >> 5);
  if (tile >= n_tiles) return;

  const int mrow  = lane & 15;
  const int khalf = lane >> 4;                 // 0 -> K pair {0,1}, 1 -> {2,3}
  int row = tile * 16 + mrow;
  if (row >= n_nodes) row = n_nodes - 1;       // clamp loads; stores guarded
  const float* hrow = h + row * 64;

  v2f a[16];
#pragma unroll
  for (int kk = 0; kk < 16; ++kk) {
    a[kk].x = hrow[4 * kk + 2 * khalf + 0];
    a[kk].y = hrow[4 * kk + 2 * khalf + 1];
  }

#pragma unroll
  for (int nt = 0; nt < 5; ++nt) {
    const int n0 = nt * 16;
    const float bv = bext[n0 + mrow];
    v8f acc;
#pragma unroll
    for (int r = 0; r < 8; ++r) acc[r] = bv;
#pragma unroll
    for (int kk = 0; kk < 16; ++kk) {
      v2f b;
      b.x = wext[(4 * kk + 2 * khalf + 0) * NPACK_STRIDE + n0 + mrow];
      b.y = wext[(4 * kk + 2 * khalf + 1) * NPACK_STRIDE + n0 + mrow];
      acc = __builtin_amdgcn_wmma_f32_16x16x4_f32(
          false, a[kk], false, b, (short)0, acc, false, false);
    }
    const int mbase = tile * 16 + 8 * khalf;
#pragma unroll
    for (int r = 0; r < 8; ++r) {
      const int m = mbase + r;
      if (m < n_nodes) npack[m * NPACK_STRIDE + n0 + mrow] = acc[r];
    }
  }
}

// ---------------- Init: out = 0, esum = 0, amax = -inf ----------------------
__global__ void init_kernel(float* __restrict__ out, float* __restrict__ amax,
                            float* __restrict__ esum, int n_out, int n_heads_tot) {
  const int i = blockIdx.x * blockDim.x + threadIdx.x;
  if (i < n_out) out[i] = 0.f;
  if (i < n_heads_tot) {
    amax[i] = -__builtin_huge_valf();
    esum[i] = 0.f;
  }
}

__device__ __forceinline__ void atomic_max_f32(float* addr, float val) {
  // int-ordering trick: valid across mixed signs with -inf init.
  if (val >= 0.f) atomicMax((int*)addr, __float_as_int(val));
  else            atomicMin((unsigned int*)addr, __float_as_uint(val));
}

// ---------------- Edge pass 1: sign, alpha, segment max ---------------------
__global__ __launch_bounds__(256) void edge_pass1_kernel(
    const int* __restrict__ src, const int* __restrict__ dst,
    const float* __restrict__ npack, const float* __restrict__ cconst,
    const float* __restrict__ ba, float* __restrict__ sign_buf,
    float* __restrict__ alpha, float* __restrict__ amax, int n_edges) {
  const int e = blockIdx.x * blockDim.x + threadIdx.x;
  if (e >= n_edges) return;
  const int s = src[e], d = dst[e];
  const float score = npack[s * NPACK_STRIDE + 64] + npack[d * NPACK_STRIDE + 65] + cconst[0];
  const float sg = (score > 0.f) ? 1.f : ((score < 0.f) ? -1.f : 0.f);
  sign_buf[e] = sg;
  const float4 p = *(const float4*)(npack + s * NPACK_STRIDE + 68);
  const float4 q = *(const float4*)(npack + d * NPACK_STRIDE + 72);
  const float bav = ba[0];
  float al[4];
  al[0] = sg * p.x + q.x + bav;
  al[1] = sg * p.y + q.y + bav;
  al[2] = sg * p.z + q.z + bav;
  al[3] = sg * p.w + q.w + bav;
#pragma unroll
  for (int hd = 0; hd < 4; ++hd) {
    const float x = al[hd];
    const float lr = (x > 0.f) ? x : 0.01f * x;   // leaky_relu slope 0.01
    alpha[e * 4 + hd] = lr;
    atomic_max_f32(&amax[d * 4 + hd], lr);
  }
}

// ---------------- Edge pass 2: ex = exp(alpha - amax), segment sum ----------
__global__ __launch_bounds__(256) void edge_pass2_kernel(
    const int* __restrict__ dst, float* __restrict__ alpha,
    const float* __restrict__ amax, float* __restrict__ esum, int n_edges) {
  const int e = blockIdx.x * blockDim.x + threadIdx.x;
  if (e >= n_edges) return;
  const int d = dst[e];
  const float4 am = *(const float4*)(amax + d * 4);
  float4 al = *(float4*)(alpha + e * 4);
  al.x = __expf(al.x - am.x);
  al.y = __expf(al.y - am.y);
  al.z = __expf(al.z - am.z);
  al.w = __expf(al.w - am.w);
  *(float4*)(alpha + e * 4) = al;   // alpha now holds ex
  atomicAdd(&esum[d * 4 + 0], al.x);
  atomicAdd(&esum[d * 4 + 1], al.y);
  atomicAdd(&esum[d * 4 + 2], al.z);
  atomicAdd(&esum[d * 4 + 3], al.w);
}

// ---------------- Edge pass 3: out[dst] += (ex/esum)*sign*hp[src] -----------
// One wave per edge; lane handles feats lane and lane+32 (coalesced 128B).
__global__ __launch_bounds__(256) void edge_pass3_kernel(
    const int* __restrict__ src, const int* __restrict__ dst,
    const float* __restrict__ npack, const float* __restrict__ sign_buf,
    const float* __restrict__ ex, const float* __restrict__ esum,
    float* __restrict__ out, int n_edges) {
  const int lane = threadIdx.x & 31;
  const int e = blockIdx.x * 8 + (threadIdx.x >> 5);
  if (e >= n_edges) return;
  const int s = src[e], d = dst[e];
  const float sg = sign_buf[e];
  const float4 exv = *(const float4*)(ex + e * 4);
  const float4 esv = *(const float4*)(esum + d * 4);
  const float w0 = sg * exv.x / esv.x;
  const float w1 = sg * exv.y / esv.y;
  const float w2 = sg * exv.z / esv.z;
  const float w3 = sg * exv.w / esv.w;
  const float hp0 = npack[s * NPACK_STRIDE + lane];
  const float hp1 = npack[s * NPACK_STRIDE + 32 + lane];
  const int hi = lane >> 4;                   // head within half
  const float c0 = hi ? w1 : w0;              // feats  0..31 -> heads 0,1
  const float c1 = hi ? w3 : w2;              // feats 32..63 -> heads 2,3
  atomicAdd(&out[d * 64 + lane],      c0 * hp0);
  atomicAdd(&out[d * 64 + 32 + lane], c1 * hp1);
}

// ---------------------------------------------------------------------------
extern "C" void kernel_launch(void* const* d_in, const int* in_sizes, int n_in,
                              void* d_out, int out_size, void* d_ws, size_t ws_size,
                              hipStream_t stream) {
  const float* h   = (const float*)d_in[0];
  const int*   src = (const int*)d_in[1];
  const int*   dst = (const int*)d_in[2];
  const float* Ww  = (const float*)d_in[3];
  const float* bw  = (const float*)d_in[4];
  const float* Wa  = (const float*)d_in[5];
  const float* ba  = (const float*)d_in[6];
  const float* Wd  = (const float*)d_in[7];
  const float* bd  = (const float*)d_in[8];
  const float* Wf  = (const float*)d_in[9];
  const float* bf  = (const float*)d_in[10];
  float* out = (float*)d_out;

  const int N = in_sizes[0] / 64;
  const int E = in_sizes[1];

  // Workspace layout (floats); float4 reads of amax/esum are 16B aligned only if E % 4 == 0.
  float* ws      = (float*)d_ws;
  float* wext    = ws;                              // 64*80
  float* bext    = ws + 8192;                       // 80
  float* cconst  = ws + 8320;                       // 1
  float* npack   = ws + 8448;                       // N*80
  float* alpha   = npack + (size_t)N * NPACK_STRIDE;// E*4  (becomes ex in pass2)
  float* signb   = alpha + (size_t)E * 4;           // E
  float* amax    = signb + (size_t)E;               // N*4
  float* esum    = amax + (size_t)N * 4;            // N*4

  const int n_tiles = (N + 15) / 16;

  precompute_kernel<<<1, 64, 0, stream>>>(Ww, bw, Wa, Wd, bd, Wf, bf,
                                          wext, bext, cconst);

  node_gemm_kernel<<<(n_tiles + 7) / 8, 256, 0, stream>>>(
      h, wext, bext, npack, N, n_tiles);

  const int n_out = N * 64;
  init_kernel<<<(n_out + 255) / 256, 256, 0, stream>>>(out, amax, esum, n_out, N * 4);

  edge_pass1_kernel<<<(E + 255) / 256, 256, 0, stream>>>(
      src, dst, npack, cconst, ba, signb, alpha, amax, E);

  edge_pass2_kernel<<<(E + 255) / 256, 256, 0, stream>>>(dst, alpha, amax, esum, E);

  edge_pass3_kernel<<<(E + 7) / 8, 256, 0, stream>>>(
      src, dst, npack, signb, alpha, esum, out, E);
}